// RadiusInteractionGraph_48163763257860
// MI455X (gfx1250) — compile-verified
//
#include <hip/hip_runtime.h>

// ---------------------------------------------------------------------------
// RadiusInteractionGraph on MI455X (gfx1250):
//   B=8192 graphs x M=64 atoms, K=32 neighbors, cutoff r=0.5.
//   - Stage 1: async global->LDS copy of positions (ASYNCcnt path) if the
//     toolchain exposes the gfx1250 builtin, else plain loads.
//   - Stage 2: 64x64 gram matrix per graph via V_WMMA_F32_16X16X4_F32
//     (16 tiles of 16x16, K=4 with z in K=2 and K=3 zero-padded);
//     sq = n2_i + n2_j - 2*g, diag -> +inf, written to LDS (stride 65).
//   - Stage 3: rank-based top-32 per node (total order == jax top_k order),
//     results scattered to LDS staging.
//   - Stage 4: fully-coalesced b128 writeout (kernel is write-BW bound:
//     ~270 MB @ 23.3 TB/s ~ 12 us; store coalescing is the only lever).
// ---------------------------------------------------------------------------

typedef __attribute__((ext_vector_type(2))) float v2f;
typedef __attribute__((ext_vector_type(8))) float v8f;

typedef __attribute__((address_space(1))) int glob_i32;   // global AS pointer elem
typedef __attribute__((address_space(3))) int lds_i32;    // LDS AS pointer elem

#define NUM_GRAPHS 8192
#define ATOMS      64
#define K_NBR      32
#define EDGES_PB   (ATOMS * K_NBR)   // 2048 edges per graph/block
#define SQ_STRIDE  65                // pad: lanes 16-31 write rows +8 at same col

#if defined(__has_builtin)
#if __has_builtin(__builtin_amdgcn_global_load_async_to_lds_b32)
#define HAVE_ASYNC_LDS 1
#endif
#endif

__device__ __forceinline__ void wait_asynccnt0() {
#if defined(__has_builtin)
#if __has_builtin(__builtin_amdgcn_s_wait_asynccnt)
    __builtin_amdgcn_s_wait_asynccnt(0);
    return;
#else
    asm volatile("s_wait_asynccnt 0x0" ::: "memory");
    return;
#endif
#else
    asm volatile("s_wait_asynccnt 0x0" ::: "memory");
#endif
}

__global__ __launch_bounds__(256) void radius_graph_kernel(
    const float* __restrict__ pos,      // (B*64, 3) row-major f32
    int*   __restrict__ out_row,        // E int32 (edge source)
    int*   __restrict__ out_col,        // E int32 (edge target)
    float* __restrict__ out_w,          // E f32   (edge weight)
    float* __restrict__ out_valid)      // E f32   (1.0 valid / 0.0 pad)
{
    __shared__ float s_xyz[ATOMS * 3];          // raw interleaved positions
    __shared__ float s_px[ATOMS];
    __shared__ float s_py[ATOMS];
    __shared__ float s_pz[ATOMS];
    __shared__ float s_n2[ATOMS];
    __shared__ float s_sq[ATOMS * SQ_STRIDE];   // 16.6 KB
    __shared__ int   st_row[EDGES_PB];          // 8 KB  output staging
    __shared__ float st_w[EDGES_PB];            // 8 KB
    __shared__ float st_valid[EDGES_PB];        // 8 KB

    const int g    = blockIdx.x;
    const int tid  = threadIdx.x;
    const int wave = tid >> 5;      // 8 waves (wave32)
    const int lane = tid & 31;

    // ---- Stage 1: positions -> LDS (async DMA path when available) ---------
#if HAVE_ASYNC_LDS
    if (tid < ATOMS * 3) {
        glob_i32* gsrc = (glob_i32*)(uintptr_t)
            (pos + (size_t)g * (ATOMS * 3) + tid);
        lds_i32* ldst = (lds_i32*)(unsigned int)(uintptr_t)
            (&s_xyz[tid]);
        __builtin_amdgcn_global_load_async_to_lds_b32(gsrc, ldst, 0, 0);
    }
    wait_asynccnt0();
#else
    if (tid < ATOMS * 3) {
        s_xyz[tid] = pos[(size_t)g * (ATOMS * 3) + tid];
    }
#endif
    __syncthreads();

    if (tid < ATOMS) {
        float x = s_xyz[3 * tid + 0];
        float y = s_xyz[3 * tid + 1];
        float z = s_xyz[3 * tid + 2];
        s_px[tid] = x;
        s_py[tid] = y;
        s_pz[tid] = z;
        s_n2[tid] = x * x + y * y + z * z;
    }
    __syncthreads();

    // ---- Stage 2: gram via WMMA f32 16x16x4, then sq matrix into LDS -------
    // ISA f32 16x4 A layout: lanes 0-15 hold {K0,K1}, lanes 16-31 hold {K2,K3};
    // B 4x16 mirrors. D: VGPR v -> row v (lanes 0-15) / row v+8 (lanes 16-31).
    {
        const int half = lane >> 4;     // 0: K={0,1}=(x,y), 1: K={2,3}=(z,0)
        const int li   = lane & 15;
#pragma unroll
        for (int u = 0; u < 2; ++u) {
            const int t  = wave + 8 * u;    // tiles: wave, wave+8
            const int mi = t >> 2, ni = t & 3;
            const int m  = mi * 16 + li;    // A row
            const int n  = ni * 16 + li;    // B col
            v2f a, b;
            a.x = half ? s_pz[m] : s_px[m];
            a.y = half ? 0.0f    : s_py[m];
            b.x = half ? s_pz[n] : s_px[n];
            b.y = half ? 0.0f    : s_py[n];
            v8f c = {};
            // (neg_a, A, neg_b, B, c_mod, C, reuse_a, reuse_b)
            v8f d = __builtin_amdgcn_wmma_f32_16x16x4_f32(
                false, a, false, b, (short)0, c, false, false);

            const float n2c = s_n2[n];
#pragma unroll
            for (int v = 0; v < 8; ++v) {
                const int r = mi * 16 + v + 8 * half;
                float s = fmaxf(s_n2[r] + n2c - 2.0f * d[v], 0.0f);
                if (r == n) s = __builtin_inff();   // mask self-loop
                s_sq[r * SQ_STRIDE + n] = s;
            }
        }
    }
    __syncthreads();

    // ---- Stage 3: per-node rank-based top-32 selection into LDS staging ----
    // Lane l owns candidates (2l, 2l+1). rank(c) = #{ j : key_j < key_c or
    // (key_j == key_c and j < c) } is a total order -> ranks are a permutation
    // of 0..63; ranks < 32 fill each output slot exactly once in ascending-
    // distance order (identical to jax.lax.top_k on -sq, ties by low index).
    for (int it = 0; it < 8; ++it) {
        const int i = wave * 8 + it;                  // local node
        const float* row = &s_sq[i * SQ_STRIDE];
        const int j0 = 2 * lane, j1 = 2 * lane + 1;
        const float k0 = row[j0], k1 = row[j1];
        int r0 = 0, r1 = 0;
#pragma unroll 8
        for (int j = 0; j < ATOMS; ++j) {
            const float kj = row[j];                  // wave-uniform broadcast
            r0 += (kj < k0) || ((kj == k0) && (j < j0));
            r1 += (kj < k1) || ((kj == k1) && (j < j1));
        }
        const int gi = g * ATOMS + i;                 // global target node

        if (r0 < K_NBR) {
            const bool ok = (k0 <= 0.25f);            // sq <= r^2
            const int  e  = i * K_NBR + r0;
            st_row[e]   = ok ? (g * ATOMS + j0) : gi;
            st_w[e]     = ok ? __builtin_sqrtf(k0) : 0.0f;
            st_valid[e] = ok ? 1.0f : 0.0f;
        }
        if (r1 < K_NBR) {
            const bool ok = (k1 <= 0.25f);
            const int  e  = i * K_NBR + r1;
            st_row[e]   = ok ? (g * ATOMS + j1) : gi;
            st_w[e]     = ok ? __builtin_sqrtf(k1) : 0.0f;
            st_valid[e] = ok ? 1.0f : 0.0f;
        }
    }
    __syncthreads();

    // ---- Stage 4: coalesced b128 writeout ----------------------------------
    // 2048 edges/block; 2 sweeps x (256 threads x 4 elems) contiguous.
    {
        const size_t base = (size_t)g * EDGES_PB;
#pragma unroll
        for (int u = 0; u < 2; ++u) {
            const int q = u * 1024 + tid * 4;         // 16B-aligned
            *(int4*)  (&out_row[base + q]) = *(const int4*)  (&st_row[q]);
            *(float4*)(&out_w[base + q])   = *(const float4*)(&st_w[q]);
            *(float4*)(&out_valid[base+q]) = *(const float4*)(&st_valid[q]);
            const int colv = g * ATOMS + (q >> 5);    // col = target node id
            *(int4*)  (&out_col[base + q]) = make_int4(colv, colv, colv, colv);
        }
    }
}

extern "C" void kernel_launch(void* const* d_in, const int* in_sizes, int n_in,
                              void* d_out, int out_size, void* d_ws, size_t ws_size,
                              hipStream_t stream) {
    (void)in_sizes; (void)n_in; (void)out_size; (void)d_ws; (void)ws_size;

    const float* pos = (const float*)d_in[0];   // d_in[1] = batch (implied by layout)

    const size_t E = (size_t)NUM_GRAPHS * ATOMS * K_NBR;   // 16,777,216 edges
    // d_out flat layout in reference return order:
    //   edge_index row [0,E) | edge_index col [E,2E) | weight [2E,3E) | valid [3E,4E)
    int*   out_row   = (int*)d_out;
    int*   out_col   = out_row + E;
    float* out_w     = (float*)d_out + 2 * E;
    float* out_valid = (float*)d_out + 3 * E;

    radius_graph_kernel<<<NUM_GRAPHS, 256, 0, stream>>>(
        pos, out_row, out_col, out_w, out_valid);
}